// TaskAlignedAssigner_34978213658776
// MI455X (gfx1250) — compile-verified
//
#include <hip/hip_runtime.h>

typedef __attribute__((ext_vector_type(2))) float v2f;
typedef __attribute__((ext_vector_type(8))) float v8f;

#define CLS   80
#define KTOP  13
#define EPSF  1e-9f

__device__ __forceinline__ bool better(float v, int i, float bv, int bi) {
  return (v > bv) || (v == bv && i < bi);
}

__device__ __forceinline__ float iou_f(float4 g, float4 b) {
  float x1 = fmaxf(g.x, b.x), y1 = fmaxf(g.y, b.y);
  float x2 = fminf(g.z, b.z), y2 = fminf(g.w, b.w);
  float inter = fmaxf(x2 - x1, 0.0f) * fmaxf(y2 - y1, 0.0f);
  float a1 = fmaxf(g.z - g.x, 0.0f) * fmaxf(g.w - g.y, 0.0f);
  float a2 = fmaxf(b.z - b.x, 0.0f) * fmaxf(b.w - b.y, 0.0f);
  return inter / (a1 + a2 - inter + EPSF);
}

// ---------------------------------------------------------------------------
// Kernel 0: init per-prior count / first-row arrays
// ---------------------------------------------------------------------------
__global__ void k_init(int* __restrict__ cnt, int* __restrict__ firstn, int P) {
  int p = blockIdx.x * 256 + threadIdx.x;
  if (p < P) { cnt[p] = 0; firstn[p] = 0x7fffffff; }
}

// ---------------------------------------------------------------------------
// Kernel 1: fused  S = onehot(label) @ cls_pred^T  (f32 WMMA, exact gather)
//           * iou^6, with streaming per-row top-13 over a prior chunk.
// grid = (NCHUNK, N/16), block = 256 (8 waves; 1 wave = one 16x16 tile)
// ---------------------------------------------------------------------------
__global__ __launch_bounds__(256) void k_metrics_topk(
    const float* __restrict__ cls_pred, const float* __restrict__ bbox_pred,
    const int* __restrict__ label_gt, const float* __restrict__ bbox_gt,
    float* __restrict__ cand_v, int* __restrict__ cand_i,
    int P, int tiles_per_chunk, int nchunk)
{
  __shared__ float4 sgt[16];
  __shared__ float  met[16][129];   // padded: avoids 16-way bank conflicts
  __shared__ float  rv[256];
  __shared__ int    ri[256];
  __shared__ float  bw_v[16];
  __shared__ int    bw_i[16];

  const int tid  = threadIdx.x;
  const int wave = tid >> 5;
  const int lane = tid & 31;
  const int g0   = blockIdx.y * 16;

  if (tid < 16) sgt[tid] = ((const float4*)bbox_gt)[g0 + tid];
  const int labm = label_gt[g0 + (lane & 15)] - 1;   // class index for A one-hot
  __syncthreads();

  const int chunk_tile0 = blockIdx.x * tiles_per_chunk;
  const int total_tiles = (P + 15) >> 4;
  int tiles_here = total_tiles - chunk_tile0;
  if (tiles_here < 0) tiles_here = 0;
  if (tiles_here > tiles_per_chunk) tiles_here = tiles_per_chunk;
  const int rounds     = (tiles_per_chunk + 7) >> 3;
  const int chunk_base = chunk_tile0 * 16;

  const int kbase_off = (lane >> 4) << 1;   // A/B K-offset: 0 (lanes 0-15) / 2 (16-31)
  const int colL      = lane & 15;
  const int rowbase   = (lane >> 4) << 3;   // C rows 0-7 / 8-15

  // A fragments (one-hot) are loop-invariant: build once, keep in registers.
  v2f areg[20];
#pragma unroll
  for (int kk = 0; kk < 20; ++kk) {
    const int base = kk * 4 + kbase_off;
    areg[kk].x = (labm == base)     ? 1.0f : 0.0f;
    areg[kk].y = (labm == base + 1) ? 1.0f : 0.0f;
  }

  // per-thread sorted top-13 (descending value, index-ascending on ties)
  float tv[KTOP]; int ti[KTOP];
#pragma unroll
  for (int k = 0; k < KTOP; ++k) { tv[k] = -__builtin_inff(); ti[k] = 0x7fffffff; }

  for (int round = 0; round < rounds; ++round) {
    int valid_tiles = tiles_here - round * 8;
    if (valid_tiles < 0) valid_tiles = 0;
    if (valid_tiles > 8) valid_tiles = 8;

    if (wave < valid_tiles) {                 // wave-uniform: EXEC all-ones for WMMA
      const int t  = round * 8 + wave;
      const int p0 = chunk_base + t * 16;
      int pb = p0 + colL;
      int pc = (pb < P) ? pb : (P - 1);
      const float* bp = cls_pred + (long)pc * CLS + kbase_off;
      __builtin_prefetch(cls_pred + (long)((pc + 128 < P) ? pc + 128 : pc) * CLS, 0, 3);

      // Preload all 20 B fragments first: one load clause + one wait,
      // instead of load->wait->wmma serialization per step.
      v2f breg[20];
#pragma unroll
      for (int kk = 0; kk < 20; ++kk) breg[kk] = *(const v2f*)(bp + kk * 4);

      v8f c = {};
#pragma unroll
      for (int kk = 0; kk < 20; ++kk) {       // K = 80 = 20 x 4, back-to-back WMMAs
        c = __builtin_amdgcn_wmma_f32_16x16x4_f32(
                false, areg[kk], false, breg[kk], (short)0, c, false, false);
      }

      const float4 pbox = ((const float4*)bbox_pred)[pc];
#pragma unroll
      for (int r = 0; r < 8; ++r) {
        const int row = rowbase + r;
        const float io = iou_f(sgt[row], pbox);
        const float i2 = io * io;
        met[row][wave * 16 + colL] = c[r] * (i2 * i2 * i2);
      }
    }
    __syncthreads();

    {   // consume: thread (row = tid&15, j = tid>>4) scans cols j, j+16, ...
      const int row = tid & 15, j = tid >> 4;
      const int ncols = valid_tiles * 16;
      const int pbase = chunk_base + round * 128;
      for (int col = j; col < ncols; col += 16) {
        const int p = pbase + col;
        if (p < P) {
          const float v = met[row][col];
          if (v > tv[KTOP - 1]) {             // strict > keeps earliest index on ties
            tv[KTOP - 1] = v; ti[KTOP - 1] = p;
#pragma unroll
            for (int q = KTOP - 1; q > 0; --q) {
              if (tv[q] > tv[q - 1]) {
                float fv = tv[q]; tv[q] = tv[q - 1]; tv[q - 1] = fv;
                int   fi = ti[q]; ti[q] = ti[q - 1]; ti[q - 1] = fi;
              }
            }
          }
        }
      }
    }
    __syncthreads();
  }

  // in-block merge: 16 sorted lists per row -> this chunk's top-13 per row
  const int row = tid & 15, j = tid >> 4;
  float* outv = cand_v + (long)(g0 + row) * (nchunk * KTOP) + blockIdx.x * KTOP;
  int*   outi = cand_i + (long)(g0 + row) * (nchunk * KTOP) + blockIdx.x * KTOP;
  unsigned used = 0;
  for (int r = 0; r < KTOP; ++r) {
    float cv = -__builtin_inff(); int ci = 0x7fffffff; int cq = -1;
#pragma unroll
    for (int q = 0; q < KTOP; ++q)
      if (!(used & (1u << q)) && better(tv[q], ti[q], cv, ci)) { cv = tv[q]; ci = ti[q]; cq = q; }
    rv[tid] = cv; ri[tid] = ci;
    __syncthreads();
    if (j == 0) {
      float bv = rv[row]; int bi = ri[row];
      for (int q = 1; q < 16; ++q) {
        float qv = rv[row + q * 16]; int qi = ri[row + q * 16];
        if (better(qv, qi, bv, bi)) { bv = qv; bi = qi; }
      }
      bw_v[row] = bv; bw_i[row] = bi;
      outv[r] = bv; outi[r] = bi;
    }
    __syncthreads();
    if (cq >= 0 && ci == bw_i[row] && cv == bw_v[row]) used |= 1u << cq;
    __syncthreads();
  }
}

// ---------------------------------------------------------------------------
// Kernel 2: per-GT-row merge of all chunk candidates -> global top-13,
//           keep-gate (max > eps), scatter count / first-row atomics.
// grid = N, block = 256
// ---------------------------------------------------------------------------
__global__ __launch_bounds__(256) void k_merge_rows(
    const float* __restrict__ cand_v, const int* __restrict__ cand_i,
    int* __restrict__ cnt, int* __restrict__ firstn, int cand_per_row)
{
  __shared__ float rv[256];
  __shared__ int   ri[256];
  __shared__ float winv[KTOP];
  __shared__ int   wini[KTOP];
  const int n = blockIdx.x, tid = threadIdx.x;
  const float* cv = cand_v + (long)n * cand_per_row;
  const int*   ci = cand_i + (long)n * cand_per_row;

  float lv[4]; int li[4];
#pragma unroll
  for (int q = 0; q < 4; ++q) { lv[q] = -__builtin_inff(); li[q] = 0x7fffffff; }
  for (int q = tid; q < cand_per_row; q += 256) {
    float v = cv[q]; int i = ci[q];
    if (better(v, i, lv[3], li[3])) {
      lv[3] = v; li[3] = i;
#pragma unroll
      for (int s = 3; s > 0; --s) {
        if (better(lv[s], li[s], lv[s - 1], li[s - 1])) {
          float fv = lv[s]; lv[s] = lv[s - 1]; lv[s - 1] = fv;
          int   fi = li[s]; li[s] = li[s - 1]; li[s - 1] = fi;
        }
      }
    }
  }

  unsigned used = 0;
  for (int r = 0; r < KTOP; ++r) {
    float bv = -__builtin_inff(); int bi = 0x7fffffff; int bq = -1;
#pragma unroll
    for (int q = 0; q < 4; ++q)
      if (!(used & (1u << q)) && better(lv[q], li[q], bv, bi)) { bv = lv[q]; bi = li[q]; bq = q; }
    rv[tid] = bv; ri[tid] = bi;
    __syncthreads();
    for (int s = 128; s >= 1; s >>= 1) {
      if (tid < s && better(rv[tid + s], ri[tid + s], rv[tid], ri[tid])) {
        rv[tid] = rv[tid + s]; ri[tid] = ri[tid + s];
      }
      __syncthreads();
    }
    float wv = rv[0]; int wi = ri[0];
    if (tid == 0) { winv[r] = wv; wini[r] = wi; }
    if (bq >= 0 && bi == wi && bv == wv) used |= 1u << bq;
    __syncthreads();
  }

  if (tid == 0 && winv[0] > EPSF) {
    for (int k = 0; k < KTOP; ++k) {
      atomicAdd(&cnt[wini[k]], 1);
      atomicMin(&firstn[wini[k]], n);
    }
  }
}

// ---------------------------------------------------------------------------
// Kernel 3: per-prior assignment (argmax-IoU only where count>1) + outputs
// ---------------------------------------------------------------------------
__global__ __launch_bounds__(256) void k_assign(
    const float* __restrict__ bbox_pred, const int* __restrict__ label_gt,
    const float* __restrict__ bbox_gt, const int* __restrict__ cnt,
    const int* __restrict__ firstn, int* __restrict__ gcls,
    float* __restrict__ out, int P, int N)
{
  __shared__ float4 sgt[256];
  __shared__ int    slbl[256];
  const int tid = threadIdx.x;
  if (tid < N) { sgt[tid] = ((const float4*)bbox_gt)[tid]; slbl[tid] = label_gt[tid]; }
  __syncthreads();

  const int p = blockIdx.x * 256 + tid;
  if (p >= P) return;
  const int c = cnt[p];
  int g = 0;
  if (c == 1) {
    g = firstn[p];
  } else if (c > 1) {           // replace with is_max_iou one-hot
    const float4 pb = ((const float4*)bbox_pred)[p];
    float bestv = -__builtin_inff(); int bestn = 0;
    for (int n = 0; n < N; ++n) {
      float io = iou_f(sgt[n], pb);
      if (io > bestv) { bestv = io; bestn = n; }   // first-occurrence argmax
    }
    g = bestn;
  }
  const int lab = slbl[g];
  out[p]     = (float)g;                              // assigned_gt_index
  out[P + p] = (c > 0) ? (float)lab : 0.0f;           // assigned_labels
  ((float4*)(out + (long)82 * P))[p] = sgt[g];        // assigned_bboxes
  gcls[p] = lab - 1;                                  // class idx for one-hot kernel
}

// ---------------------------------------------------------------------------
// Kernel 4: coalesced one-hot class output (P x 80 floats)
// block = 320 threads = 4 priors x 80 classes
// ---------------------------------------------------------------------------
__global__ __launch_bounds__(320) void k_onehot(
    const int* __restrict__ gcls, float* __restrict__ out, int P)
{
  const int tid = threadIdx.x;
  const int p = blockIdx.x * 4 + tid / CLS;
  const int cc = tid % CLS;
  if (p < P) out[(long)2 * P + (long)p * CLS + cc] = (gcls[p] == cc) ? 1.0f : 0.0f;
}

// ---------------------------------------------------------------------------
extern "C" void kernel_launch(void* const* d_in, const int* in_sizes, int n_in,
                              void* d_out, int out_size, void* d_ws, size_t ws_size,
                              hipStream_t stream)
{
  const float* cls_pred  = (const float*)d_in[0];
  const float* bbox_pred = (const float*)d_in[1];
  const int*   label_gt  = (const int*)d_in[2];
  const float* bbox_gt   = (const float*)d_in[3];
  float* out = (float*)d_out;

  const int P = in_sizes[0] / CLS;   // 150000
  const int N = in_sizes[2];         // 256

  const int NCHUNK = 64;
  const int total_tiles     = (P + 15) / 16;
  const int tiles_per_chunk = (total_tiles + NCHUNK - 1) / NCHUNK;
  const int cand_per_row    = NCHUNK * KTOP;

  char* ws = (char*)d_ws;
  float* cand_v = (float*)ws;  ws += (size_t)N * cand_per_row * sizeof(float);
  int*   cand_i = (int*)ws;    ws += (size_t)N * cand_per_row * sizeof(int);
  int*   cnt    = (int*)ws;    ws += (size_t)P * sizeof(int);
  int*   firstn = (int*)ws;    ws += (size_t)P * sizeof(int);
  int*   gcls   = (int*)ws;    // + P*4  (total ~3.5 MB)

  k_init<<<(P + 255) / 256, 256, 0, stream>>>(cnt, firstn, P);

  dim3 g1(NCHUNK, N / 16);
  k_metrics_topk<<<g1, 256, 0, stream>>>(cls_pred, bbox_pred, label_gt, bbox_gt,
                                         cand_v, cand_i, P, tiles_per_chunk, NCHUNK);

  k_merge_rows<<<N, 256, 0, stream>>>(cand_v, cand_i, cnt, firstn, cand_per_row);

  k_assign<<<(P + 255) / 256, 256, 0, stream>>>(bbox_pred, label_gt, bbox_gt,
                                                cnt, firstn, gcls, out, P, N);

  k_onehot<<<(P + 3) / 4, 320, 0, stream>>>(gcls, out, P);
}